// GNN_23072564314645
// MI455X (gfx1250) — compile-verified
//
#include <hip/hip_runtime.h>
#include <math.h>

typedef float v2f __attribute__((ext_vector_type(2)));
typedef float v8f __attribute__((ext_vector_type(8)));

constexpr int NN = 20000;   // nodes
constexpr int EE = 320000;  // edges
constexpr int DD = 512;     // input feature dim
constexpr int HH = 512;     // hidden dim

// ---------------- degree / symmetric norm ----------------
__global__ void k_deg_init(float* __restrict__ deg) {
  int i = blockIdx.x * blockDim.x + threadIdx.x;
  if (i < NN) deg[i] = 1.0f;  // self-loop contributes 1
}

__global__ void k_deg_count(const int* __restrict__ dst, float* __restrict__ deg) {
  int e = blockIdx.x * blockDim.x + threadIdx.x;
  if (e < EE) atomicAdd(&deg[dst[e]], 1.0f);
}

__global__ void k_dinv(const float* __restrict__ deg, float* __restrict__ dinv) {
  int i = blockIdx.x * blockDim.x + threadIdx.x;
  if (i < NN) dinv[i] = rsqrtf(deg[i]);
}

// ---------------- pack W rows into K/2 x Nc float2 tiles ----------------
// Wp2[kp*Nc + n] = (W[2kp][n], W[2kp+1][n]) -> one b64 load per B fragment.
__global__ void k_packW(const float* __restrict__ W, float2* __restrict__ Wp,
                        int K, int Nc) {
  int idx = blockIdx.x * blockDim.x + threadIdx.x;
  if (idx >= (K / 2) * Nc) return;
  int kp = idx / Nc;
  int n  = idx - kp * Nc;
  Wp[idx] = make_float2(W[(size_t)(2 * kp) * Nc + n],
                        W[(size_t)(2 * kp + 1) * Nc + n]);
}

// ---------------- dense GEMM via f32 WMMA ----------------
// C[M x Nc] = A[M x K] * Wp2 (packed B), row-major fp32.
// grid.x = M/32 row tiles. Block = (32,4): each wave owns a 32x128 C tile
// (16 x v8f accumulators), wave y covers columns [128*y, 128*y+128).
// Ping-pong fragment buffers (no copy rotation), pointer-increment addressing.
// WMMA f32 16x16x4 fragment layouts (ISA 7.12.2):
//   A (16x4):  lane = M + 16*(K>=2), vgpr k&1  -> v2f per lane
//   B (4x16):  lane = N + 16*(K>=2), vgpr k&1  -> v2f per lane (= Wp2[kp][n])
//   C/D 16x16: vgpr r -> M = r + 8*(lane>=16), N = lane&15

struct Frag {
  v2f a0, a1;   // A rows [0,16) and [16,32) of the 32-row tile
  v2f b[8];     // 8 column tiles
};

__device__ __forceinline__ void load_frag(Frag& f, const float* ap0,
                                          const float* ap1, const float2* bp) {
  f.a0 = *(const v2f*)ap0;
  f.a1 = *(const v2f*)ap1;
#pragma unroll
  for (int j = 0; j < 8; ++j) {
    float2 t = bp[16 * j];
    f.b[j] = (v2f){t.x, t.y};
  }
}

__device__ __forceinline__ void do_wmma(v8f* acc, const Frag& f) {
#pragma unroll
  for (int j = 0; j < 8; ++j) {
    acc[j] = __builtin_amdgcn_wmma_f32_16x16x4_f32(
        false, f.a0, false, f.b[j], (short)0, acc[j], false, false);
    acc[8 + j] = __builtin_amdgcn_wmma_f32_16x16x4_f32(
        false, f.a1, false, f.b[j], (short)0, acc[8 + j], false, false);
  }
}

__global__ __launch_bounds__(128) void k_gemm_wmma(const float* __restrict__ A,
                                                   const float2* __restrict__ Wp2,
                                                   float* __restrict__ C,
                                                   int K, int Nc) {
  const int lane = threadIdx.x;
  const int l16  = lane & 15;
  const int half = lane >> 4;  // 0 or 1
  const int rowBase = blockIdx.x * 32;
  const int colBase = threadIdx.y * 128;

  v8f acc[16] = {};

  // pointer-increment addressing: one k-step = +4 floats (A), +2*Nc float2 (B)
  const float*  ap0 = A + (size_t)(rowBase + l16) * K + 2 * half;
  const float*  ap1 = ap0 + (size_t)16 * K;
  const float2* bp  = Wp2 + (size_t)half * Nc + colBase + l16;
  const size_t  bstep = 2 * (size_t)Nc;

  Frag fA, fB;

  // prologue: k-step 0 into fA
  load_frag(fA, ap0, ap1, bp);
  ap0 += 4; ap1 += 4; bp += bstep;

  // steady state: 2 k-steps per iteration, fully ping-ponged
  const int iters = K / 8 - 1;  // K % 8 == 0
  for (int it = 0; it < iters; ++it) {
    load_frag(fB, ap0, ap1, bp);
    ap0 += 4; ap1 += 4; bp += bstep;
    do_wmma(acc, fA);
    load_frag(fA, ap0, ap1, bp);
    ap0 += 4; ap1 += 4; bp += bstep;
    do_wmma(acc, fB);
  }

  // tail: last k-step
  load_frag(fB, ap0, ap1, bp);
  do_wmma(acc, fA);
  do_wmma(acc, fB);

#pragma unroll
  for (int rt = 0; rt < 2; ++rt) {
#pragma unroll
    for (int j = 0; j < 8; ++j) {
#pragma unroll
      for (int r = 0; r < 8; ++r) {
        const int m = rowBase + rt * 16 + r + 8 * half;
        C[(size_t)m * Nc + colBase + 16 * j + l16] = acc[rt * 8 + j][r];
      }
    }
  }
}

// ---------------- edge scatter-add: agg[dst] += h[src] * dinv[src]*dinv[dst] ----------------
__global__ void k_scatter(const float* __restrict__ h, const int* __restrict__ src,
                          const int* __restrict__ dst, const float* __restrict__ dinv,
                          float* __restrict__ agg) {
  const int e = blockIdx.x;
  const int s = src[e];
  const int d = dst[e];
  const float w = dinv[s] * dinv[d];
  const float* hs = h + (size_t)s * HH;
  float* od = agg + (size_t)d * HH;
  for (int f = threadIdx.x; f < HH; f += blockDim.x)
    atomicAdd(&od[f], hs[f] * w);
}

// ---------------- epilogue: self-loop + bias + ReLU (in place on agg) ----------------
__global__ void k_finalize(float* __restrict__ agg, const float* __restrict__ h,
                           const float* __restrict__ dinv, const float* __restrict__ bias) {
  const size_t i = (size_t)blockIdx.x * blockDim.x + threadIdx.x;
  if (i >= (size_t)NN * HH) return;
  const int node = (int)(i >> 9);  // HH == 512
  const int f = (int)(i & 511);
  const float di = dinv[node];
  const float v = agg[i] + h[i] * di * di + bias[f];
  agg[i] = v > 0.0f ? v : 0.0f;
}

// ---------------- column-wise max pool over all nodes ----------------
// x >= 0 after ReLU, so uint-bit-pattern atomicMax is order preserving.
__global__ void k_maxpool(const float* __restrict__ x, float* __restrict__ out) {
  const int f = blockIdx.x * blockDim.x + threadIdx.x;  // column
  const int ROWS = 500;                                 // NN / gridDim.y
  const int r0 = blockIdx.y * ROWS;
  float m = 0.0f;
  for (int r = 0; r < ROWS; ++r) m = fmaxf(m, x[(size_t)(r0 + r) * HH + f]);
  atomicMax((unsigned int*)&out[f], __float_as_uint(m));
}

extern "C" void kernel_launch(void* const* d_in, const int* in_sizes, int n_in,
                              void* d_out, int out_size, void* d_ws, size_t ws_size,
                              hipStream_t stream) {
  (void)in_sizes; (void)n_in; (void)out_size; (void)ws_size;

  const float* feat = (const float*)d_in[0];   // [NN, DD]
  const int*   eidx = (const int*)d_in[1];     // [2, EE]
  // d_in[2] = edge_attr, unused by GCNConv
  const float* W1 = (const float*)d_in[3];     // [DD, HH]
  const float* b1 = (const float*)d_in[4];     // [HH]
  const float* W2 = (const float*)d_in[5];     // [HH, HH]
  const float* b2 = (const float*)d_in[6];     // [HH]
  const int* srcI = eidx;
  const int* dstI = eidx + EE;

  // workspace layout
  float*  deg  = (float*)d_ws;                       // [NN]
  float*  dinv = deg + NN;                           // [NN]
  float2* W1p  = (float2*)(dinv + NN);               // [(DD/2)*HH]
  float2* W2p  = W1p + (size_t)(DD / 2) * HH;        // [(HH/2)*HH]
  float*  bufH = (float*)(W2p + (size_t)(HH / 2) * HH);  // [NN*HH] GEMM out (h)
  float*  bufG = bufH + (size_t)NN * HH;                 // [NN*HH] agg / activations

  const size_t featBytes = (size_t)NN * HH * sizeof(float);

  // --- symmetric normalization coefficients + weight packing ---
  k_deg_init <<<(NN + 255) / 256, 256, 0, stream>>>(deg);
  k_deg_count<<<(EE + 255) / 256, 256, 0, stream>>>(dstI, deg);
  k_dinv     <<<(NN + 255) / 256, 256, 0, stream>>>(deg, dinv);
  k_packW<<<((DD / 2) * HH + 255) / 256, 256, 0, stream>>>(W1, W1p, DD, HH);
  k_packW<<<((HH / 2) * HH + 255) / 256, 256, 0, stream>>>(W2, W2p, HH, HH);

  // --- layer 1 ---
  k_gemm_wmma<<<dim3(NN / 32), dim3(32, 4), 0, stream>>>(feat, W1p, bufH, DD, HH);
  hipMemsetAsync(bufG, 0, featBytes, stream);
  k_scatter  <<<EE, 256, 0, stream>>>(bufH, srcI, dstI, dinv, bufG);
  k_finalize <<<(NN * HH + 255) / 256, 256, 0, stream>>>(bufG, bufH, dinv, b1);
  // bufG = x1 (ReLU'd), bufH free

  // --- layer 2 ---
  k_gemm_wmma<<<dim3(NN / 32), dim3(32, 4), 0, stream>>>(bufG, W2p, bufH, HH, HH);
  hipMemsetAsync(bufG, 0, featBytes, stream);  // stream-ordered after GEMM2 reads x1
  k_scatter  <<<EE, 256, 0, stream>>>(bufH, srcI, dstI, dinv, bufG);
  k_finalize <<<(NN * HH + 255) / 256, 256, 0, stream>>>(bufG, bufH, dinv, b2);
  // bufG = x2

  // --- global max pool -> [1, HH] ---
  hipMemsetAsync(d_out, 0, HH * sizeof(float), stream);
  k_maxpool<<<dim3(HH / 256, NN / 500), 256, 0, stream>>>(bufG, (float*)d_out);
}